// TRGNN_25546465477054
// MI455X (gfx1250) — compile-verified
//
#include <hip/hip_runtime.h>
#include <math.h>

#define TT 4
#define BB 8
#define NN 25000
#define LL 64
#define NCH 32

typedef __attribute__((ext_vector_type(2))) float v2f;
typedef __attribute__((ext_vector_type(8))) float v8f;

static __device__ inline v8f wmma4(v2f a, v2f b, v8f c) {
    // V_WMMA_F32_16X16X4_F32 : D(16x16,f32) = A(16x4,f32) * B(4x16,f32) + C
    return __builtin_amdgcn_wmma_f32_16x16x4_f32(
        false, a, false, b, (short)0, c, false, false);
}

__global__ void zero_kernel(float* p, unsigned long long n) {
    unsigned long long i = (unsigned long long)blockIdx.x * 256ull + threadIdx.x;
    if (i < n) p[i] = 0.f;
}

// ---------------------------------------------------------------------------
// Per-step GRU update of node memory using f32 WMMA (16x16x4).
// Each wave owns a 16-node tile. LDS stages enc / msg / old mem tiles.
// ---------------------------------------------------------------------------
__global__ __launch_bounds__(128) void gru_kernel(
    const float* __restrict__ t, const float* __restrict__ raw,
    const float* __restrict__ w_t, const float* __restrict__ b_t,
    const float* __restrict__ W_msg, const float* __restrict__ b_msg,
    const float* __restrict__ Wi, const float* __restrict__ Wh,
    const float* __restrict__ bi, const float* __restrict__ bh,
    const float* __restrict__ r_last, float* __restrict__ mem,
    int step, int ntiles, int wavesTotal)
{
    __shared__ float s_enc[4][16 * 64];
    __shared__ float s_msg[4][16 * 64];
    __shared__ float s_mem[4][16 * 64];

    const int lane = threadIdx.x & 31;
    const int wv   = threadIdx.x >> 5;
    const int gw   = blockIdx.x * 4 + wv;
    const int half = lane >> 4;      // 0: lanes 0-15, 1: lanes 16-31
    const int l16  = lane & 15;
    const int iters = (ntiles + wavesTotal - 1) / wavesTotal;

    for (int it = 0; it < iters; ++it) {
        int tile = gw + it * wavesTotal;
        const bool active = (tile < ntiles);
        if (!active) tile = ntiles - 1;       // clamp; predicated stores below
        const int nb = tile * 16;

        // ---- stage enc(t of batch B-1) and old memory tile into LDS ----
        for (int idx = lane; idx < 16 * 64; idx += 32) {
            int m = idx >> 6, l = idx & 63;
            int node = nb + m; if (node >= NN) node = NN - 1;
            float tv = t[((size_t)step * BB + (BB - 1)) * (size_t)NN + node];
            s_enc[wv][idx] = cosf(tv * w_t[l] + b_t[l]);
            s_mem[wv][idx] = mem[(size_t)node * 64 + l];
        }
        __syncthreads();

        // ---- msg = b_msg + r_prev * W_msg[0,:] + enc @ W_msg[1:,:] ----
        for (int cb = 0; cb < 4; ++cb) {
            v8f acc = {};
            const int col = cb * 16 + l16;
            for (int kb = 0; kb < 16; ++kb) {
                int kk = kb * 4 + half * 2;
                v2f a; a[0] = s_enc[wv][(l16 << 6) + kk];
                       a[1] = s_enc[wv][(l16 << 6) + kk + 1];
                v2f bf; bf[0] = W_msg[(size_t)(1 + kk) * 64 + col];
                        bf[1] = W_msg[(size_t)(2 + kk) * 64 + col];
                acc = wmma4(a, bf, acc);
            }
            const float wm0 = W_msg[col];
            const float bm  = b_msg[col];
            for (int v = 0; v < 8; ++v) {
                int m = v + half * 8;
                int node = nb + m; if (node >= NN) node = NN - 1;
                float rp = (step == 0) ? raw[(size_t)(BB - 1) * NN + node]
                                       : r_last[node];
                s_msg[wv][(m << 6) + col] = acc[v] + bm + rp * wm0;
            }
        }
        __syncthreads();

        // ---- GRU: gi = msg@Wi+bi, gh = mem@Wh+bh (6 gate blocks per cb) ----
        for (int cb = 0; cb < 4; ++cb) {
            v8f air = {}, aiz = {}, ain = {}, ahr = {}, ahz = {}, ahn = {};
            const int col = cb * 16 + l16;
            for (int kb = 0; kb < 16; ++kb) {
                int kk = kb * 4 + half * 2;
                v2f ax; ax[0] = s_msg[wv][(l16 << 6) + kk];
                        ax[1] = s_msg[wv][(l16 << 6) + kk + 1];
                v2f ah; ah[0] = s_mem[wv][(l16 << 6) + kk];
                        ah[1] = s_mem[wv][(l16 << 6) + kk + 1];
                const float* wi0 = Wi + (size_t)kk * 192;
                const float* wi1 = Wi + (size_t)(kk + 1) * 192;
                const float* wh0 = Wh + (size_t)kk * 192;
                const float* wh1 = Wh + (size_t)(kk + 1) * 192;
                v2f bir, biz, bin, bhr, bhz, bhn;
                bir[0] = wi0[col];       bir[1] = wi1[col];
                biz[0] = wi0[64 + col];  biz[1] = wi1[64 + col];
                bin[0] = wi0[128 + col]; bin[1] = wi1[128 + col];
                bhr[0] = wh0[col];       bhr[1] = wh1[col];
                bhz[0] = wh0[64 + col];  bhz[1] = wh1[64 + col];
                bhn[0] = wh0[128 + col]; bhn[1] = wh1[128 + col];
                air = wmma4(ax, bir, air);
                aiz = wmma4(ax, biz, aiz);
                ain = wmma4(ax, bin, ain);
                ahr = wmma4(ah, bhr, ahr);
                ahz = wmma4(ah, bhz, ahz);
                ahn = wmma4(ah, bhn, ahn);
            }
            const float biR = bi[col], biZ = bi[64 + col], biN = bi[128 + col];
            const float bhR = bh[col], bhZ = bh[64 + col], bhN = bh[128 + col];
            for (int v = 0; v < 8; ++v) {
                int m = v + half * 8;
                int node = nb + m;
                float ir = air[v] + biR, hr = ahr[v] + bhR;
                float iz = aiz[v] + biZ, hz = ahz[v] + bhZ;
                float in_ = ain[v] + biN, hn = ahn[v] + bhN;
                float rg = 1.f / (1.f + expf(-(ir + hr)));
                float zg = 1.f / (1.f + expf(-(iz + hz)));
                float ng = tanhf(in_ + rg * hn);
                float hold = s_mem[wv][(m << 6) + col];
                float outv = (1.f - zg) * ng + zg * hold;
                if (active && node < NN)
                    mem[(size_t)node * 64 + col] = outv;
            }
        }
        __syncthreads();
    }
}

// ---------------------------------------------------------------------------
// Per-(step,b): q = tar_h @ Wq + bq ; qk = Wk q (129) ; cdot = q . bk
// ---------------------------------------------------------------------------
__global__ __launch_bounds__(160) void qk_kernel(
    const float* __restrict__ raw, const float* __restrict__ r_last,
    const float* __restrict__ mem, const int* __restrict__ tar,
    const float* __restrict__ b_t, const float* __restrict__ Wq,
    const float* __restrict__ bq, const float* __restrict__ Wk,
    const float* __restrict__ bk, float* __restrict__ qk, int step)
{
    __shared__ float sq[64];
    const int b = blockIdx.x;
    const int tid = threadIdx.x;
    const int tb = tar[step * BB + b];
    if (tid < 64) {
        float tr = (step == 0) ? raw[(size_t)b * NN + tb] : r_last[tb];
        float acc = bq[tid] + tr * Wq[tid];
        for (int l = 0; l < 64; ++l)
            acc += mem[(size_t)tb * 64 + l] * Wq[(size_t)(1 + l) * 64 + tid];
        for (int l = 0; l < 64; ++l)
            acc += cosf(b_t[l]) * Wq[(size_t)(65 + l) * 64 + tid];
        sq[tid] = acc;
    }
    __syncthreads();
    if (tid < 129) {
        float acc = 0.f;
        for (int j = 0; j < 64; ++j) acc += Wk[(size_t)tid * 64 + j] * sq[j];
        qk[b * 132 + tid] = acc;
    }
    if (tid == 129) {
        float c = 0.f;
        for (int j = 0; j < 64; ++j) c += sq[j] * bk[j];
        qk[b * 132 + 129] = c;
    }
}

// ---------------------------------------------------------------------------
// Pass 1: score[b,n] = scale*(h . qk + q.bk), masked; chunk max
// wave-per-node: lanes split the 129 dims (fully coalesced mem reads)
// ---------------------------------------------------------------------------
__global__ __launch_bounds__(256) void score_kernel(
    const float* __restrict__ t, const int* __restrict__ n_mask,
    const float* __restrict__ raw, const float* __restrict__ r_last,
    const float* __restrict__ mem, const float* __restrict__ w_t,
    const float* __restrict__ b_t, const float* __restrict__ qk,
    float* __restrict__ scores, float* __restrict__ cmax, int step)
{
    __shared__ float smax[8];
    const int b = blockIdx.y, chunk = blockIdx.x;
    const int lane = threadIdx.x & 31, wv = threadIdx.x >> 5;
    const float* qkb = qk + b * 132;
    const float q_r = qkb[0], cdot = qkb[129];
    const float qm0 = qkb[1 + lane],  qm1 = qkb[33 + lane];
    const float qe0 = qkb[65 + lane], qe1 = qkb[97 + lane];
    const float wt0 = w_t[lane], wt1 = w_t[lane + 32];
    const float bt0 = b_t[lane], bt1 = b_t[lane + 32];
    const size_t tb = ((size_t)step * BB + b) * (size_t)NN;
    float mx = -3.0e38f;
    for (int n = chunk * 8 + wv; n < NN; n += NCH * 8) {
        float tv = t[tb + n];
        float p = qm0 * mem[(size_t)n * 64 + lane]
                + qm1 * mem[(size_t)n * 64 + 32 + lane]
                + qe0 * cosf(tv * wt0 + bt0)
                + qe1 * cosf(tv * wt1 + bt1);
        for (int off = 16; off; off >>= 1) p += __shfl_xor(p, off, 32);
        float rbn = (step == 0) ? raw[(size_t)b * NN + n] : r_last[n];
        float sc = (p + q_r * rbn + cdot) * 0.125f;
        sc = (n_mask[tb + n] > 0) ? sc : -1.0e9f;
        if (lane == 0) scores[(size_t)b * NN + n] = sc;
        mx = fmaxf(mx, sc);
    }
    if (lane == 0) smax[wv] = mx;
    __syncthreads();
    if (threadIdx.x == 0) {
        float m2 = smax[0];
        for (int i = 1; i < 8; ++i) m2 = fmaxf(m2, smax[i]);
        cmax[b * NCH + chunk] = m2;
    }
}

__global__ void reduce_kernel(const float* __restrict__ cmax,
                              float* __restrict__ gmax, float* __restrict__ hb)
{
    const int b = blockIdx.x, lane = threadIdx.x;
    float m = -3.0e38f;
    for (int i = lane; i < NCH; i += 32) m = fmaxf(m, cmax[b * NCH + i]);
    for (int off = 16; off; off >>= 1) m = fmaxf(m, __shfl_xor(m, off, 32));
    if (lane == 0) gmax[b] = m;
    for (int i = lane; i < 132; i += 32) hb[b * 132 + i] = 0.f;
}

// ---------------------------------------------------------------------------
// Pass 2: e = exp(score-max); hbar += e*h ; sum += e   (LDS then global f32 atomics)
// ---------------------------------------------------------------------------
__global__ __launch_bounds__(256) void accum_kernel(
    const float* __restrict__ t, const float* __restrict__ raw,
    const float* __restrict__ r_last, const float* __restrict__ mem,
    const float* __restrict__ w_t, const float* __restrict__ b_t,
    const float* __restrict__ scores, const float* __restrict__ gmax,
    float* __restrict__ hb, int step)
{
    __shared__ float sh[132];
    for (int i = threadIdx.x; i < 132; i += 256) sh[i] = 0.f;
    __syncthreads();
    const int b = blockIdx.y, chunk = blockIdx.x;
    const int lane = threadIdx.x & 31, wv = threadIdx.x >> 5;
    const float wt0 = w_t[lane], wt1 = w_t[lane + 32];
    const float bt0 = b_t[lane], bt1 = b_t[lane + 32];
    const float gm = gmax[b];
    const size_t tb = ((size_t)step * BB + b) * (size_t)NN;
    float a0 = 0.f, a1 = 0.f, a2 = 0.f, a3 = 0.f, ar = 0.f, as = 0.f;
    for (int n = chunk * 8 + wv; n < NN; n += NCH * 8) {
        float e = expf(scores[(size_t)b * NN + n] - gm);
        float tv = t[tb + n];
        a0 += e * mem[(size_t)n * 64 + lane];
        a1 += e * mem[(size_t)n * 64 + 32 + lane];
        a2 += e * cosf(tv * wt0 + bt0);
        a3 += e * cosf(tv * wt1 + bt1);
        if (lane == 0) {
            float rbn = (step == 0) ? raw[(size_t)b * NN + n] : r_last[n];
            ar += e * rbn; as += e;
        }
    }
    atomicAdd(&sh[1 + lane],  a0);
    atomicAdd(&sh[33 + lane], a1);
    atomicAdd(&sh[65 + lane], a2);
    atomicAdd(&sh[97 + lane], a3);
    if (lane == 0) { atomicAdd(&sh[0], ar); atomicAdd(&sh[129], as); }
    __syncthreads();
    for (int i = threadIdx.x; i < 130; i += 256) atomicAdd(&hb[b * 132 + i], sh[i]);
}

__global__ __launch_bounds__(64) void finalize_kernel(
    const float* __restrict__ hb, const float* __restrict__ Wv,
    const float* __restrict__ bv, const float* __restrict__ W_out,
    const float* __restrict__ b_out, float* __restrict__ out, int step)
{
    __shared__ float sz[64];
    const int b = blockIdx.x, j = threadIdx.x;
    const float* h = hb + b * 132;
    const float inv = 1.f / h[129];
    float acc = bv[j];
    for (int d = 0; d < 129; ++d) acc += (h[d] * inv) * Wv[(size_t)d * 64 + j];
    sz[j] = acc * W_out[j];
    __syncthreads();
    if (j == 0) {
        float s = b_out[0];
        for (int i = 0; i < 64; ++i) s += sz[i];
        out[step * BB + b] = s;
    }
}

__global__ void rupdate_kernel(
    const float* __restrict__ raw, const float* __restrict__ r_label,
    const int* __restrict__ tar, const float* __restrict__ out,
    float* __restrict__ r_last, int step)
{
    const int n = blockIdx.x * 256 + threadIdx.x;
    if (n >= NN) return;
    float v = (step == 0) ? raw[(size_t)(BB - 1) * NN + n] : r_last[n];
    for (int b = 0; b < BB; ++b)                 // last-b wins (scatter set)
        if (tar[step * BB + b] == n) v = out[step * BB + b];
    r_last[n] = 0.5f * (v + r_label[((size_t)step * BB + (BB - 1)) * NN + n]);
}

extern "C" void kernel_launch(void* const* d_in, const int* in_sizes, int n_in,
                              void* d_out, int out_size, void* d_ws, size_t ws_size,
                              hipStream_t stream) {
    const float* raw     = (const float*)d_in[0];
    const float* r_label = (const float*)d_in[1];
    const float* t       = (const float*)d_in[2];
    const int*   tar     = (const int*)d_in[4];
    const int*   n_mask  = (const int*)d_in[5];
    const float* w_t     = (const float*)d_in[6];
    const float* b_t     = (const float*)d_in[7];
    const float* W_msg   = (const float*)d_in[8];
    const float* b_msg   = (const float*)d_in[9];
    const float* Wi      = (const float*)d_in[10];
    const float* Wh      = (const float*)d_in[11];
    const float* bi      = (const float*)d_in[12];
    const float* bh      = (const float*)d_in[13];
    const float* Wq      = (const float*)d_in[14];
    const float* bq      = (const float*)d_in[15];
    const float* Wk      = (const float*)d_in[16];
    const float* bk      = (const float*)d_in[17];
    const float* Wv      = (const float*)d_in[18];
    const float* bv      = (const float*)d_in[19];
    const float* W_out   = (const float*)d_in[20];
    const float* b_out   = (const float*)d_in[21];
    float* out = (float*)d_out;

    float* ws     = (float*)d_ws;
    float* mem    = ws;                          // N*64
    float* r_last = mem + (size_t)NN * 64;       // N
    float* scores = r_last + NN;                 // B*N
    float* qk     = scores + (size_t)BB * NN;    // B*132
    float* hb     = qk + BB * 132;               // B*132
    float* gmax   = hb + BB * 132;               // B
    float* cmax   = gmax + BB;                   // B*NCH

    zero_kernel<<<((size_t)NN * 64 + 255) / 256, 256, 0, stream>>>(mem, (size_t)NN * 64);

    const int ntiles = (NN + 15) / 16;           // 1563
    const int gruBlocks = 196;                   // 784 waves -> 2 tiles/wave
    const int wavesTotal = gruBlocks * 4;

    for (int s = 0; s < TT; ++s) {
        gru_kernel<<<gruBlocks, 128, 0, stream>>>(t, raw, w_t, b_t, W_msg, b_msg,
                                                  Wi, Wh, bi, bh, r_last, mem,
                                                  s, ntiles, wavesTotal);
        qk_kernel<<<BB, 160, 0, stream>>>(raw, r_last, mem, tar, b_t, Wq, bq,
                                          Wk, bk, qk, s);
        score_kernel<<<dim3(NCH, BB), 256, 0, stream>>>(t, n_mask, raw, r_last,
                                                        mem, w_t, b_t, qk,
                                                        scores, cmax, s);
        reduce_kernel<<<BB, 32, 0, stream>>>(cmax, gmax, hb);
        accum_kernel<<<dim3(NCH, BB), 256, 0, stream>>>(t, raw, r_last, mem,
                                                        w_t, b_t, scores, gmax,
                                                        hb, s);
        finalize_kernel<<<BB, 64, 0, stream>>>(hb, Wv, bv, W_out, b_out, out, s);
        rupdate_kernel<<<(NN + 255) / 256, 256, 0, stream>>>(raw, r_label, tar,
                                                             out, r_last, s);
    }
}